// NeighborAggregator_86277303042418
// MI455X (gfx1250) — compile-verified
//
#include <hip/hip_runtime.h>
#include <cstdint>
#include <cstddef>

// ---------------------------------------------------------------------------
// NeighborAggregator for MI455X (gfx1250):
//   reduced_sum = rowsum(adj * data)   (memory-bound: 512 MB @ 23.3 TB/s)
//   alpha       = softmax(reduced_sum)
// Phase 1 streams both matrices with GLOBAL_LOAD_ASYNC_TO_LDS_B128 (double
// buffered, ASYNCcnt-paced) and folds per-lane partials into 16 row sums with
// a single V_WMMA_F32_16X16X4_F32 (A = partials in the native 16x4 fragment
// layout, B = ones) per wave. Phase 2 is one deterministic block.
// ---------------------------------------------------------------------------

#define NN 8192

constexpr int ROWS   = 16;            // rows per wave tile (WMMA M)
constexpr int CHUNK  = 32;            // columns per streamed chunk
constexpr int PITCH  = CHUNK + 4;     // LDS row pitch in floats (144 B, 16B-aligned)
constexpr int WPB    = 4;             // waves per block
constexpr int SPLITS = 4;             // column splits per row tile
constexpr int COLS_PER_WAVE = NN / SPLITS;        // 2048
constexpr int NCHUNKS       = COLS_PER_WAVE / CHUNK; // 64

typedef float v2f __attribute__((ext_vector_type(2)));
typedef float v8f __attribute__((ext_vector_type(8)));

__device__ __forceinline__ unsigned lds_off(const void* p) {
  // low 32 bits of a flat pointer into LDS == byte offset within the group segment
  return (unsigned)(uintptr_t)p;
}

// Copy a 16-row x 128-byte (32-float) tile from global (row stride NN floats)
// into LDS rows of PITCH floats. 4 async instructions x 512 B, fully coalesced.
__device__ __forceinline__ void async_copy_tile(const float* __restrict__ gsrc,
                                                float* lbuf, int lane) {
#pragma unroll
  for (int i = 0; i < 4; ++i) {
    const int f   = i * 32 + lane;
    const int row = f >> 3;     // 8 lanes (8 x 16B = 128B) per row
    const int seg = f & 7;
    const float* g = gsrc + (size_t)row * NN + (size_t)seg * 4;
    const unsigned l = lds_off(&lbuf[row * PITCH + seg * 4]);
    asm volatile("global_load_async_to_lds_b128 %0, %1, off"
                 :: "v"(l), "v"((uint64_t)(uintptr_t)g)
                 : "memory");
  }
}

__global__ void __launch_bounds__(WPB * 32)
na_rowsum_kernel(const float* __restrict__ data,
                 const float* __restrict__ adj,
                 float* __restrict__ partial) {
  __shared__ __align__(16) float lds[WPB][2][2][ROWS * PITCH];

  const int lane = threadIdx.x & 31;
  const int w    = threadIdx.x >> 5;
  const int gw   = blockIdx.x * WPB + w;   // 0 .. 2047
  const int tile  = gw >> 2;               // 0 .. 511
  const int split = gw & 3;                // 0 .. 3

  const size_t rowBase = (size_t)tile * ROWS;
  const size_t colBase = (size_t)split * COLS_PER_WAVE;
  const float* dbase = data + rowBase * NN + colBase;
  const float* abase = adj  + rowBase * NN + colBase;

  // Prefetch chunk 0 into buffer 0.
  async_copy_tile(dbase, &lds[w][0][0][0], lane);
  async_copy_tile(abase, &lds[w][0][1][0], lane);

  // Lane roles for the WMMA A-fragment: lanes m and m+16 both own row m.
  const int p = lane >> 4;   // which K-slot pair / which column half
  const int m = lane & 15;   // row within tile
  float acc0 = 0.0f, acc1 = 0.0f;

  for (int c = 0; c < NCHUNKS; ++c) {
    const int cur = c & 1;
    if (c + 1 < NCHUNKS) {
      const int nxt = (c + 1) & 1;
      async_copy_tile(dbase + (size_t)(c + 1) * CHUNK, &lds[w][nxt][0][0], lane);
      async_copy_tile(abase + (size_t)(c + 1) * CHUNK, &lds[w][nxt][1][0], lane);
      // 8 newest (next chunk) may remain outstanding; previous 8 must be done.
      asm volatile("s_wait_asynccnt 8" ::: "memory");
    } else {
      asm volatile("s_wait_asynccnt 0" ::: "memory");
    }

    const float* ld = &lds[w][cur][0][m * PITCH + p * 16];
    const float* la = &lds[w][cur][1][m * PITCH + p * 16];
#pragma unroll
    for (int j = 0; j < 4; ++j) {
      const float4 d4 = *(const float4*)(ld + j * 4);
      const float4 a4 = *(const float4*)(la + j * 4);
      acc0 = fmaf(d4.x, a4.x, acc0);
      acc0 = fmaf(d4.y, a4.y, acc0);
      acc1 = fmaf(d4.z, a4.z, acc1);
      acc1 = fmaf(d4.w, a4.w, acc1);
    }
  }

  // Fold 4 partials per row (lanes m / m+16, two regs each) into row sums:
  // A = partials laid out as the native 16x4 f32 fragment, B = ones =>
  // D[m][n] = sum_k A[m][k] = full row-m sum in every column.
  v2f A;  A[0] = acc0; A[1] = acc1;
  v2f Bv; Bv[0] = 1.0f; Bv[1] = 1.0f;
  v8f Cv = {0.f, 0.f, 0.f, 0.f, 0.f, 0.f, 0.f, 0.f};
  v8f D = __builtin_amdgcn_wmma_f32_16x16x4_f32(
      /*neg_a=*/false, A, /*neg_b=*/false, Bv,
      /*c_mod=*/(short)0, Cv, /*reuse_a=*/false, /*reuse_b=*/false);

  // D layout: VGPR r, lanes 0-15 -> M=r ; lanes 16-31 -> M=8+r.
  if (m == 0) {  // lanes 0 and 16 hold column N=0 of all 16 rows
    float* dst = partial + ((size_t)tile * SPLITS + split) * ROWS + p * 8;
#pragma unroll
    for (int r = 0; r < 8; ++r) dst[r] = D[r];
  }
}

__global__ void __launch_bounds__(1024)
na_softmax_kernel(const float* __restrict__ partial, float* __restrict__ out) {
  __shared__ float red[1024];
  const int tid = threadIdx.x;

  float vals[8];
  float lmax = -INFINITY;
#pragma unroll
  for (int k = 0; k < 8; ++k) {
    const int r    = tid * 8 + k;
    const int tile = r >> 4;
    const int mm   = r & 15;
    const float* ptr = partial + (size_t)tile * (SPLITS * ROWS) + mm;
    // deterministic fixed-order combine of the 4 column splits
    const float s = ((ptr[0] + ptr[16]) + ptr[32]) + ptr[48];
    out[NN + r] = s;          // reduced_sum
    vals[k] = s;
    lmax = fmaxf(lmax, s);
  }

  red[tid] = lmax;
  __syncthreads();
  for (int s = 512; s > 0; s >>= 1) {
    if (tid < s) red[tid] = fmaxf(red[tid], red[tid + s]);
    __syncthreads();
  }
  const float gmax = red[0];
  __syncthreads();

  float e[8];
  float lsum = 0.0f;
#pragma unroll
  for (int k = 0; k < 8; ++k) {
    e[k] = expf(vals[k] - gmax);
    lsum += e[k];
  }
  red[tid] = lsum;
  __syncthreads();
  for (int s = 512; s > 0; s >>= 1) {
    if (tid < s) red[tid] += red[tid + s];
    __syncthreads();
  }
  const float inv = 1.0f / red[0];
#pragma unroll
  for (int k = 0; k < 8; ++k) out[tid * 8 + k] = e[k] * inv;  // alpha
}

extern "C" void kernel_launch(void* const* d_in, const int* in_sizes, int n_in,
                              void* d_out, int out_size, void* d_ws, size_t ws_size,
                              hipStream_t stream) {
  const float* data = (const float*)d_in[0];
  const float* adj  = (const float*)d_in[1];
  float* out        = (float*)d_out;
  float* partial    = (float*)d_ws;   // 512 * 4 * 16 floats = 128 KB

  const int totalWaves = (NN / ROWS) * SPLITS;   // 2048
  const int blocks     = totalWaves / WPB;       // 512

  hipLaunchKernelGGL(na_rowsum_kernel, dim3(blocks), dim3(WPB * 32), 0, stream,
                     data, adj, partial);
  hipLaunchKernelGGL(na_softmax_kernel, dim3(1), dim3(1024), 0, stream,
                     partial, out);
}